// VectorQuantizer_89008902242479
// MI455X (gfx1250) — compile-verified
//
#include <hip/hip_runtime.h>

typedef __attribute__((ext_vector_type(2))) float v2f;
typedef __attribute__((ext_vector_type(8))) float v8f;

// 16-byte vector type matching the async-LDS builtin's parameter type
typedef int v4i_t __attribute__((vector_size(16)));
typedef __attribute__((address_space(1))) v4i_t gv4i;
typedef __attribute__((address_space(3))) v4i_t lv4i;

#define VQ_N      65536        // total latent vectors (2 * 32^3)
#define VQ_D      64
#define VQ_K      1024
#define ROWS_PER_BLOCK 128     // 8 waves * 16 rows
#define LDS_PITCH 68           // 64 + pad: 16B-aligned rows, conflict-free b64 reads

// ---------------------------------------------------------------------------
// Kernel 1: codebook squared norms -> d_ws[0..1023]
// ---------------------------------------------------------------------------
__global__ void vq_enorm_kernel(const float* __restrict__ emb,
                                float* __restrict__ enorm) {
    int r = blockIdx.x * blockDim.x + threadIdx.x;
    if (r < VQ_K) {
        const float4* row = (const float4*)(emb + ((size_t)r << 6));
        float s = 0.f;
#pragma unroll
        for (int i = 0; i < 16; ++i) {
            float4 v = row[i];
            s += v.x * v.x + v.y * v.y + v.z * v.z + v.w * v.w;
        }
        enorm[r] = s;
    }
}

// ---------------------------------------------------------------------------
// Kernel 2: main — fp32 WMMA distances, async-LDS double-buffered codebook,
//           argmin, gather, loss partials
// ---------------------------------------------------------------------------
__global__ __launch_bounds__(256)
void vq_main_kernel(const float* __restrict__ z,
                    const float* __restrict__ emb,
                    const float* __restrict__ enorm,
                    float* __restrict__ out,
                    float* __restrict__ idx_out,
                    float* __restrict__ partials) {
    __shared__ float lds_e[2][64 * LDS_PITCH];  // double-buffered codebook chunk
    __shared__ int   lds_idx[ROWS_PER_BLOCK];
    __shared__ float lds_red[8];

    const int tid  = threadIdx.x;
    const int lane = tid & 31;
    const int wave = tid >> 5;
    const int mrow = lane & 15;   // C-matrix column slot / A row
    const int hh   = lane >> 4;   // lane half

    const int rowbase = blockIdx.x * ROWS_PER_BLOCK + wave * 16;
    const int n  = rowbase + mrow;
    const int b  = n >> 15;       // batch (tiles never straddle the 32768 boundary)
    const int sp = n & 32767;
    const float* zb = z + ((size_t)b << 21) + sp;   // &z[b][0][spatial]

    // A fragments: lane(mrow,hh) holds x[mrow][4k+2h], x[mrow][4k+2h+1]
    float a0[16], a1[16];
#pragma unroll
    for (int kk = 0; kk < 16; ++kk) {
        int d = kk * 4 + 2 * hh;
        a0[kk] = zb[(size_t)d << 15];
        a1[kk] = zb[(size_t)(d + 1) << 15];
    }

    float best[8];
    int   bidx[8];
#pragma unroll
    for (int r = 0; r < 8; ++r) { best[r] = 3.0e38f; bidx[r] = 0; }

    // Per-thread staging slice: 4 x 16B async copies, coalesced.
    const int srow = tid >> 2, sseg = tid & 3;

    auto issue_chunk = [&](int c) {
        const float* g = emb + ((size_t)(c * 64 + srow) << 6) + sseg * 16;
        float*       l = &lds_e[c & 1][srow * LDS_PITCH + sseg * 16];
#pragma unroll
        for (int j = 0; j < 4; ++j) {
            __builtin_amdgcn_global_load_async_to_lds_b128(
                (gv4i*)(g + j * 4), (lv4i*)(l + j * 4), 0, 0);
        }
    };

    issue_chunk(0);

    for (int c = 0; c < 16; ++c) {          // 16 chunks of 64 codebook columns
        __builtin_amdgcn_s_wait_asynccnt(0);   // my chunk-c copies done
        __syncthreads();                       // everyone's copies done / prev reads done
        if (c + 1 < 16) issue_chunk(c + 1);    // overlap chunk c+1 copy with compute

        const float* ebuf = &lds_e[c & 1][0];
#pragma unroll
        for (int tt = 0; tt < 4; ++tt) {    // four 16-col tiles per chunk
            v8f acc = {};
            const int colL = tt * 16 + mrow;
#pragma unroll
            for (int kk = 0; kk < 16; ++kk) {
                int d = kk * 4 + 2 * hh;
                v2f bfrag = *(const v2f*)&ebuf[colL * LDS_PITCH + d];
                v2f afrag; afrag.x = a0[kk]; afrag.y = a1[kk];
                acc = __builtin_amdgcn_wmma_f32_16x16x4_f32(
                    false, afrag, false, bfrag, (short)0, acc, false, false);
            }
            const int   col = c * 64 + colL;
            const float en  = enorm[col];      // 64B coalesced, L2-hot
#pragma unroll
            for (int r = 0; r < 8; ++r) {
                float sc = en - 2.0f * acc[r];  // relative distance (||x||^2 dropped)
                if (sc < best[r]) { best[r] = sc; bidx[r] = col; }
            }
        }
    }

    // argmin reduce across the 16 lanes of each half (ties -> lowest index)
#pragma unroll
    for (int r = 0; r < 8; ++r) {
#pragma unroll
        for (int off = 1; off <= 8; off <<= 1) {
            float ov = __shfl_xor(best[r], off, 32);
            int   oi = __shfl_xor(bidx[r], off, 32);
            if (ov < best[r] || (ov == best[r] && oi < bidx[r])) {
                best[r] = ov; bidx[r] = oi;
            }
        }
    }
    if (mrow == 0) {
#pragma unroll
        for (int r = 0; r < 8; ++r)       // row m = r + 8*half
            lds_idx[wave * 16 + hh * 8 + r] = bidx[r];
    }
    __syncthreads();

    // gather z_q, write output + indices, accumulate loss partial
    const int   n2    = rowbase + mrow;
    const int   myidx = lds_idx[wave * 16 + mrow];
    const int   b2 = n2 >> 15;
    const int   s2 = n2 & 32767;
    const float* er  = emb + ((size_t)myidx << 6);
    const float* zr  = z   + ((size_t)b2 << 21) + s2;
    float*      orow = out + ((size_t)b2 << 21) + s2;
    float acc = 0.f;
#pragma unroll
    for (int dd = 0; dd < 32; ++dd) {
        int d = dd * 2 + hh;
        float ev = er[d];
        float zv = zr[(size_t)d << 15];
        orow[(size_t)d << 15] = ev;       // coalesced across lanes (fixed d, adjacent s)
        float df = ev - zv;
        acc += df * df;
    }
    if (hh == 0) idx_out[n2] = (float)myidx;

    // block-level loss reduction -> d_ws partial (non-atomic: replay-safe)
#pragma unroll
    for (int off = 16; off >= 1; off >>= 1) acc += __shfl_xor(acc, off, 32);
    if (lane == 0) lds_red[wave] = acc;
    __syncthreads();
    if (tid == 0) {
        float tot = 0.f;
#pragma unroll
        for (int w = 0; w < 8; ++w) tot += lds_red[w];
        partials[blockIdx.x] = tot;
    }
}

// ---------------------------------------------------------------------------
// Kernel 3: finalize loss = 1.25 * sum / (N*D)   (deterministic overwrite)
// ---------------------------------------------------------------------------
__global__ void vq_loss_kernel(const float* __restrict__ partials,
                               float* __restrict__ loss) {
    __shared__ float red[16];
    int tid = threadIdx.x;           // 512 threads
    float v = partials[tid];
#pragma unroll
    for (int off = 16; off >= 1; off >>= 1) v += __shfl_xor(v, off, 32);
    if ((tid & 31) == 0) red[tid >> 5] = v;
    __syncthreads();
    if (tid == 0) {
        float tot = 0.f;
#pragma unroll
        for (int i = 0; i < 16; ++i) tot += red[i];
        *loss = 1.25f * tot / 4194304.0f;   // (0.25 + 1.0) * mean sq diff
    }
}

// ---------------------------------------------------------------------------
extern "C" void kernel_launch(void* const* d_in, const int* in_sizes, int n_in,
                              void* d_out, int out_size, void* d_ws, size_t ws_size,
                              hipStream_t stream) {
    const float* z   = (const float*)d_in[0];   // [2,64,32,32,32]
    const float* emb = (const float*)d_in[1];   // [1024,64]

    float* out      = (float*)d_out;            // z_q (straight-through fwd value)
    float* loss     = out + 4194304;            // scalar
    float* idx_out  = out + 4194305;            // 65536 indices (as float)

    float* enorm    = (float*)d_ws;             // [1024]
    float* partials = enorm + VQ_K;             // [512]

    vq_enorm_kernel<<<4, 256, 0, stream>>>(emb, enorm);
    vq_main_kernel<<<VQ_N / ROWS_PER_BLOCK, 256, 0, stream>>>(
        z, emb, enorm, out, idx_out, partials);
    vq_loss_kernel<<<1, 512, 0, stream>>>(partials, loss);
}